// SelfAttention_15539191677143
// MI455X (gfx1250) — compile-verified
//
#include <hip/hip_runtime.h>
#include <hip/hip_bf16.h>

typedef __attribute__((ext_vector_type(2))) float v2f;
typedef __attribute__((ext_vector_type(8))) float v8f;
typedef __attribute__((ext_vector_type(4))) int   i32x4;

// ---------------- problem constants ----------------
#define BATCH     8
#define HW        128
#define NT        (HW * HW)      // 16384 tokens per image
#define EMB       256            // EMBED_DIM == VALUE_DIM
#define NH        8
#define VHD       32
#define BS        8
#define BS2       64
#define TOPK      16
#define NBT       256
#define KATT      (BS2 + TOPK)   // 80
#define MTOT      (BATCH * NT)   // 131072

// ---------------- async global->LDS helpers ----------------
typedef __attribute__((address_space(1))) i32x4 gi32x4;   // global int4
typedef __attribute__((address_space(3))) i32x4 li32x4;   // LDS int4

__device__ __forceinline__ void async_b128(const float* g, float* l) {
#if __has_builtin(__builtin_amdgcn_global_load_async_to_lds_b128)
    __builtin_amdgcn_global_load_async_to_lds_b128((gi32x4*)g, (li32x4*)l, 0, 0);
#else
    const float4 t = *(const float4*)g;
    *(v2f*)l       = v2f{t.x, t.y};
    *((v2f*)l + 1) = v2f{t.z, t.w};
#endif
}

__device__ __forceinline__ void wait_async() {
#if __has_builtin(__builtin_amdgcn_s_wait_asynccnt)
    __builtin_amdgcn_s_wait_asynccnt(0);
#elif __has_builtin(__builtin_amdgcn_global_load_async_to_lds_b128)
    asm volatile("s_wait_asynccnt 0" ::: "memory");
#endif
}

// ================= Kernel 1: xp[M,256] = x[M,256] @ W_in[256,256] + b_in =================
#define TM   128
#define TN   64
#define TK   16
#define LDPA 20   // A-tile stride: even (b64 frags), mult-4 (b128 stores), 20m%64 distinct
#define LDPB 72   // B-tile stride: mult-4 (b128)

__global__ __launch_bounds__(256)
void gemm_bias_kernel(const float* __restrict__ src, const float* __restrict__ Wm,
                      const float* __restrict__ bias, float* __restrict__ dst) {
    __shared__ float sA[2][TM * LDPA];
    __shared__ float sB[2][TK * LDPB];

    const int tid  = threadIdx.x;
    const int lane = tid & 31;
    const int wv   = tid >> 5;
    const int lr   = lane & 15;
    const int lh   = lane >> 4;

    const int n0b = blockIdx.x * TN;
    const int m0  = blockIdx.y * TM;
    const int m0w = wv * 16;

    float bvals[4];
#pragma unroll
    for (int nt = 0; nt < 4; ++nt) bvals[nt] = bias[n0b + nt * 16 + lr];

    const int ar0 = tid >> 2,         ac0 = (tid & 3) << 2;
    const int ar1 = (tid + 256) >> 2, ac1 = ac0;
    const int bk  = tid >> 2,         bc  = (tid & 3) << 2;

    auto stage = [&](int buf, int kb) {
        async_b128(src + (size_t)(m0 + ar0) * EMB + kb + ac0, &sA[buf][ar0 * LDPA + ac0]);
        async_b128(src + (size_t)(m0 + ar1) * EMB + kb + ac1, &sA[buf][ar1 * LDPA + ac1]);
        async_b128(Wm  + (size_t)(kb + bk) * EMB + n0b + bc,  &sB[buf][bk  * LDPB + bc]);
    };

    v8f acc[4] = {v8f{}, v8f{}, v8f{}, v8f{}};

    stage(0, 0);
    for (int it = 0; it < EMB / TK; ++it) {
        const int buf = it & 1;
        wait_async();
        __syncthreads();
        if (it + 1 < EMB / TK) stage(buf ^ 1, (it + 1) * TK);

        const float* __restrict__ pA = sA[buf];
        const float* __restrict__ pB = sB[buf];
#pragma unroll
        for (int kk = 0; kk < TK / 4; ++kk) {
            const int k0 = kk * 4 + lh * 2;
            const v2f a = *(const v2f*)&pA[(m0w + lr) * LDPA + k0];
#pragma unroll
            for (int nt = 0; nt < 4; ++nt) {
                const int n = nt * 16 + lr;
                const v2f b = v2f{pB[k0 * LDPB + n], pB[(k0 + 1) * LDPB + n]};
                acc[nt] = __builtin_amdgcn_wmma_f32_16x16x4_f32(
                    false, a, false, b, (short)0, acc[nt], false, false);
            }
        }
    }

#pragma unroll
    for (int nt = 0; nt < 4; ++nt) {
        const int col = n0b + nt * 16 + lr;
#pragma unroll
        for (int i = 0; i < 8; ++i) {
            const int row = m0 + m0w + i + 8 * lh;
            dst[(size_t)row * EMB + col] = acc[nt][i] + bvals[nt];
        }
    }
}

// ========== Kernel 2 (fused): per (batch, block):  all 8 heads of attention into
//            yL(64x256 in LDS), then out(64x256) = yL @ W_out + b_out ==========
#define LDK  84    // attn-tile / VT stride
#define LDY  260   // yL stride: 260*4%16==0 (b128), 4m%64 distinct banks, even
#define LDW  260   // W_out-tile stride
#define ATT_F (BS2 * LDK + VHD * LDK)   // 8064 floats (phase-A region)
#define WBUF_F (2 * TK * LDW)           // 8320 floats (phase-B region)
#define R2_F  (ATT_F > WBUF_F ? ATT_F : WBUF_F)

__global__ __launch_bounds__(256)
void attn_out_fused_kernel(const float* __restrict__ xp, const float* __restrict__ aw,
                           const int* __restrict__ idx, const float* __restrict__ wts,
                           const float* __restrict__ Wout, const float* __restrict__ bout,
                           float* __restrict__ out) {
    __shared__ float smem[BS2 * LDY + R2_F];
    float* yL  = smem;                 // 64 x 260
    float* r2  = smem + BS2 * LDY;     // overlapped: phase A tiles / phase B W tiles
    float* sAt = r2;                   // 64 x 84
    float* sVT = r2 + BS2 * LDK;       // 32 x 84

    const int blk = blockIdx.x;
    const int bI  = blockIdx.y;

    const int tid  = threadIdx.x;
    const int lane = tid & 31;
    const int wv   = tid >> 5;
    const int lr   = lane & 15;
    const int lh   = lane >> 4;

    const int bh = blk >> 4, bw = blk & 15;

    // ---------------- Phase A: attention, one head at a time ----------------
    const int m0a = (wv >> 1) * 16;
    const int n0a = (wv & 1) * 16;

    for (int hh = 0; hh < NH; ++hh) {
        // async copy attn tile (64x80, contiguous): 1280 b128 units
        const size_t abase = ((((size_t)hh * BATCH + bI) * NBT) + blk) * (size_t)(BS2 * KATT);
        for (int u = tid; u < BS2 * (KATT / 4); u += 256) {
            const int m = u / (KATT / 4);
            const int c = (u - m * (KATT / 4)) << 2;
            async_b128(aw + abase + (size_t)m * KATT + c, &sAt[m * LDK + c]);
        }
        // gather V (scaled, transposed): 80 x 8 float4 units
        for (int u = tid; u < KATT * (VHD / 4); u += 256) {
            const int k = u >> 3;
            const int c = (u & 7) << 2;
            int token;
            float scale = 1.0f;
            if (k < BS2) {
                token = (bh * BS + (k >> 3)) * HW + (bw * BS + (k & 7));
            } else {
                const size_t so = ((size_t)bI * NBT + blk) * TOPK + (k - BS2);
                token = idx[so];
                scale = wts[so];
            }
            const float4 t = *(const float4*)(xp + ((size_t)bI * NT + token) * EMB + hh * VHD + c);
            sVT[(c + 0) * LDK + k] = t.x * scale;
            sVT[(c + 1) * LDK + k] = t.y * scale;
            sVT[(c + 2) * LDK + k] = t.z * scale;
            sVT[(c + 3) * LDK + k] = t.w * scale;
        }
        wait_async();
        __syncthreads();

        v8f acc = v8f{};
#pragma unroll
        for (int ks = 0; ks < KATT / 4; ++ks) {
            const int k0 = ks * 4 + lh * 2;
            const v2f a = *(const v2f*)&sAt[(m0a + lr) * LDK + k0];
            const v2f b = *(const v2f*)&sVT[(n0a + lr) * LDK + k0];
            acc = __builtin_amdgcn_wmma_f32_16x16x4_f32(
                false, a, false, b, (short)0, acc, false, false);
        }
        // write O tile into yL
#pragma unroll
        for (int i = 0; i < 8; ++i) {
            const int m = m0a + i + 8 * lh;
            yL[m * LDY + hh * VHD + n0a + lr] = acc[i];
        }
        __syncthreads();   // protect sAt/sVT (and finally r2) before reuse
    }

    // ---------------- Phase B: out(64x256) = yL(64x256) @ W_out + b_out ----------------
    // wave wv -> cols [wv*32, wv*32+32) (2 n-tiles), all 4 m-tiles
    const int nw0 = wv * 32;
    float bv[2];
#pragma unroll
    for (int ct = 0; ct < 2; ++ct) bv[ct] = bout[nw0 + ct * 16 + lr];

    float* sW0 = r2;
    float* sW1 = r2 + TK * LDW;

    auto stageW = [&](float* buf, int kb) {
        // 16 rows x 64 b128 chunks = 1024 units, 4 per thread
        for (int u = tid; u < TK * (EMB / 4); u += 256) {
            const int k = u >> 6;
            const int c = (u & 63) << 2;
            async_b128(Wout + (size_t)(kb + k) * EMB + c, buf + k * LDW + c);
        }
    };

    v8f acc2[4][2];
#pragma unroll
    for (int mt = 0; mt < 4; ++mt)
#pragma unroll
        for (int ct = 0; ct < 2; ++ct) acc2[mt][ct] = v8f{};

    stageW(sW0, 0);
    for (int it = 0; it < EMB / TK; ++it) {
        wait_async();
        __syncthreads();
        float* cur = (it & 1) ? sW1 : sW0;
        float* nxt = (it & 1) ? sW0 : sW1;
        if (it + 1 < EMB / TK) stageW(nxt, (it + 1) * TK);

#pragma unroll
        for (int kk = 0; kk < TK / 4; ++kk) {
            const int k0 = kk * 4 + lh * 2;
            v2f bfr[2];
#pragma unroll
            for (int ct = 0; ct < 2; ++ct) {
                const int n = nw0 + ct * 16 + lr;
                bfr[ct] = v2f{cur[k0 * LDW + n], cur[(k0 + 1) * LDW + n]};
            }
#pragma unroll
            for (int mt = 0; mt < 4; ++mt) {
                const v2f a = *(const v2f*)&yL[(mt * 16 + lr) * LDY + it * TK + k0];
#pragma unroll
                for (int ct = 0; ct < 2; ++ct) {
                    acc2[mt][ct] = __builtin_amdgcn_wmma_f32_16x16x4_f32(
                        false, a, false, bfr[ct], (short)0, acc2[mt][ct], false, false);
                }
            }
        }
    }

    // store out: row m (block-local) -> token (bh*8 + m/8, bw*8 + m%8)
#pragma unroll
    for (int mt = 0; mt < 4; ++mt) {
#pragma unroll
        for (int i = 0; i < 8; ++i) {
            const int m   = mt * 16 + i + 8 * lh;
            const int row = bh * BS + (m >> 3);
            const int col = bw * BS + (m & 7);
            const size_t tbase = ((size_t)bI * NT + row * HW + col) * EMB;
#pragma unroll
            for (int ct = 0; ct < 2; ++ct) {
                out[tbase + nw0 + ct * 16 + lr] = acc2[mt][ct][i] + bv[ct];
            }
        }
    }
}

// ---------------- launcher ----------------
extern "C" void kernel_launch(void* const* d_in, const int* in_sizes, int n_in,
                              void* d_out, int out_size, void* d_ws, size_t ws_size,
                              hipStream_t stream) {
    const float* x     = (const float*)d_in[0];
    const float* aw    = (const float*)d_in[1];
    const int*   idx   = (const int*)d_in[2];
    const float* wts   = (const float*)d_in[3];
    const float* W_in  = (const float*)d_in[4];
    const float* b_in  = (const float*)d_in[5];
    const float* W_out = (const float*)d_in[6];
    const float* b_out = (const float*)d_in[7];

    float* xp = (float*)d_ws;

    dim3 ggrid(EMB / TN, MTOT / TM);          // (4, 1024)
    gemm_bias_kernel<<<ggrid, 256, 0, stream>>>(x, W_in, b_in, xp);

    dim3 fgrid(NBT, BATCH);                   // (256, 8)
    attn_out_fused_kernel<<<fgrid, 256, 0, stream>>>(xp, aw, idx, wts,
                                                     W_out, b_out, (float*)d_out);
}